// ExponentialLinear_23089744183382
// MI455X (gfx1250) — compile-verified
//
#include <hip/hip_runtime.h>

typedef __attribute__((ext_vector_type(2))) float v2f;
typedef __attribute__((ext_vector_type(4))) float v4f;
typedef __attribute__((ext_vector_type(8))) float v8f;
typedef int v4i __attribute__((vector_size(16)));   // matches builtin param type

#define DIM 512
#define BM 128
#define BK 16
#define LDST (BK + 4)    // A-tile LDS row stride (floats): 80B rows -> 16B aligned
#define N_TAYLOR 14      // ||A||~0.45 -> term 14 ~ 1e-16 relative; fp32-exact

#if __has_builtin(__builtin_amdgcn_global_load_async_to_lds_b128)
#define USE_ASYNC_LDS 1
#else
#define USE_ASYNC_LDS 0
#endif

// Builtin signature (from hipcc diagnostic): param0 = v4i AS1*, param1 = v4i AS3*
#define GAS4(p) ((__attribute__((address_space(1))) v4i*)(p))
#define LAS4(p) ((__attribute__((address_space(3))) v4i*)(p))

__device__ __forceinline__ void wait_async_zero() {
#if USE_ASYNC_LDS
#if __has_builtin(__builtin_amdgcn_s_wait_asynccnt)
  __builtin_amdgcn_s_wait_asynccnt(0);
#else
  asm volatile("s_wait_asynccnt 0" ::: "memory");
#endif
#endif
}

// ---------------------------------------------------------------------------
// A[i][j] = angles[triu_idx(i,j)] (i<j), -angles[triu_idx(j,i)] (i>j), 0 diag
// ---------------------------------------------------------------------------
__global__ __launch_bounds__(256) void build_skew_kernel(
    const float* __restrict__ angles, float* __restrict__ A) {
  int j = blockIdx.x * 64 + (threadIdx.x & 63);
  int i = blockIdx.y * 4 + (threadIdx.x >> 6);
  float v = 0.0f;
  if (i < j) {
    int idx = i * (DIM - 1) - (i * (i - 1)) / 2 + (j - i - 1);
    v = angles[idx];
  } else if (i > j) {
    int idx = j * (DIM - 1) - (j * (j - 1)) / 2 + (i - j - 1);
    v = -angles[idx];
  }
  A[i * DIM + j] = v;
}

// W = I + A  (Taylor terms k=0,1)
__global__ __launch_bounds__(256) void init_w_kernel(
    const float* __restrict__ A, float* __restrict__ W) {
  int t = blockIdx.x * 256 + threadIdx.x;
  int i = t / DIM, j = t % DIM;
  W[t] = A[t] + (i == j ? 1.0f : 0.0f);
}

// ---------------------------------------------------------------------------
// D = alpha*(A@B) [+ bias];  optionally accum += alpha*(A@B)
// A: [M x 512], B: [512 x 512], row-major. Block tile: BM x (32*TN).
// 8 wave32s as 4x2; each wave: 32 x (TN*16) via 2xTN f32 16x16x4 WMMA tiles.
// Double-buffered LDS; async global->LDS prefetch (ASYNCcnt) when available.
// ---------------------------------------------------------------------------
template <int TN>
__global__ __launch_bounds__(256) void gemm_f32_wmma_kernel(
    const float* __restrict__ A, const float* __restrict__ B,
    float* __restrict__ D, float alpha,
    float* __restrict__ accum, const float* __restrict__ bias) {
  constexpr int BN   = 32 * TN;       // 64 or 128
  constexpr int BSTR = BN + 4;        // B-tile LDS row stride: 16B aligned
  constexpr int BPARTS = (BK * BN) / 1024;  // float4 transfers per thread for B
  const int N = DIM, K = DIM;

  __shared__ float As[2][BM * LDST];  // [row][k]
  __shared__ float Bs[2][BK * BSTR];  // [k][col]  (async-copy friendly)

  const int tid  = threadIdx.x;
  const int lane = tid & 31;             // wave32
  const int wave = tid >> 5;             // 0..7
  const int wm   = wave >> 1;            // 0..3 : 32-row group
  const int wn   = wave & 1;             // 0..1 : (TN*16)-col group
  const int lm   = lane & 15;
  const int kh   = (lane >> 4) << 1;     // K offset 0 or 2 per ISA A/B layout

  const int blockM = blockIdx.y * BM;
  const int blockN = blockIdx.x * BN;

  // Per-thread staging coordinates
  const int a_row0 = tid >> 2;           // + r*64
  const int a_c4   = (tid & 3) << 2;
  constexpr int FPER = BN / 4;           // float4 per B row
  const int b_row0 = tid / FPER;         // + r*(256/FPER)
  const int b_c4   = (tid % FPER) << 2;

  v8f acc[2][TN] = {};

  auto stage = [&](int buf, int k0) {
#if USE_ASYNC_LDS
#pragma unroll
    for (int r = 0; r < 2; ++r) {
      int row = a_row0 + r * 64;
      __builtin_amdgcn_global_load_async_to_lds_b128(
          GAS4(A + (size_t)(blockM + row) * K + k0 + a_c4),
          LAS4(&As[buf][row * LDST + a_c4]), 0, 0);
    }
#pragma unroll
    for (int r = 0; r < BPARTS; ++r) {
      int krow = b_row0 + r * (256 / FPER);
      __builtin_amdgcn_global_load_async_to_lds_b128(
          GAS4(B + (size_t)(k0 + krow) * N + blockN + b_c4),
          LAS4(&Bs[buf][krow * BSTR + b_c4]), 0, 0);
    }
#else
    // Pipelined fallback: issue ALL global loads, then all LDS stores.
    v4f av[2], bv[BPARTS];
#pragma unroll
    for (int r = 0; r < 2; ++r)
      av[r] = *(const v4f*)(A + (size_t)(blockM + a_row0 + r * 64) * K + k0 + a_c4);
#pragma unroll
    for (int r = 0; r < BPARTS; ++r)
      bv[r] = *(const v4f*)(B + (size_t)(k0 + b_row0 + r * (256 / FPER)) * N + blockN + b_c4);
#pragma unroll
    for (int r = 0; r < 2; ++r)
      *(v4f*)&As[buf][(a_row0 + r * 64) * LDST + a_c4] = av[r];
#pragma unroll
    for (int r = 0; r < BPARTS; ++r)
      *(v4f*)&Bs[buf][(b_row0 + r * (256 / FPER)) * BSTR + b_c4] = bv[r];
#endif
  };

  stage(0, 0);
  wait_async_zero();
  __syncthreads();

  for (int k0 = 0; k0 < K; k0 += BK) {
    const int cur = (k0 / BK) & 1;
    if (k0 + BK < K) stage(cur ^ 1, k0 + BK);   // prefetch next tile

#pragma unroll
    for (int ks = 0; ks < BK; ks += 4) {
      v2f afrag[2], bfrag[TN];
#pragma unroll
      for (int t = 0; t < 2; ++t) {
        // A operand 16x4: lanes 0-15 -> (K,K+1) of row M=lm; lanes 16-31 -> (K+2,K+3)
        int row = wm * 32 + t * 16 + lm;
        afrag[t] = *(const v2f*)&As[cur][row * LDST + ks + kh];
      }
#pragma unroll
      for (int n = 0; n < TN; ++n) {
        int col = wn * (TN * 16) + n * 16 + lm;
        v2f b;
        b.x = Bs[cur][(ks + kh) * BSTR + col];
        b.y = Bs[cur][(ks + kh + 1) * BSTR + col];
        bfrag[n] = b;
      }
#pragma unroll
      for (int t = 0; t < 2; ++t)
#pragma unroll
        for (int n = 0; n < TN; ++n)
          acc[t][n] = __builtin_amdgcn_wmma_f32_16x16x4_f32(
              false, afrag[t], false, bfrag[n],
              (short)0, acc[t][n], false, false);
    }

    wait_async_zero();   // own prefetch landed in LDS
    __syncthreads();     // everyone's prefetch landed; everyone done reading cur
  }

  // Epilogue: D VGPR e holds rows e / e+8 across lane halves (ISA C/D layout)
  const int mhalf = (lane >> 4) << 3;
#pragma unroll
  for (int t = 0; t < 2; ++t) {
#pragma unroll
    for (int n = 0; n < TN; ++n) {
      int col = blockN + wn * (TN * 16) + n * 16 + lm;
      float bv = bias ? bias[col] : 0.0f;
#pragma unroll
      for (int e = 0; e < 8; ++e) {
        int row = blockM + wm * 32 + t * 16 + mhalf + e;
        float v = acc[t][n][e] * alpha;
        size_t idx = (size_t)row * N + col;
        if (D) D[idx] = v + bv;
        if (accum) accum[idx] += v;   // one writer per element -> race-free
      }
    }
  }
}

// ---------------------------------------------------------------------------
extern "C" void kernel_launch(void* const* d_in, const int* in_sizes, int n_in,
                              void* d_out, int out_size, void* d_ws, size_t ws_size,
                              hipStream_t stream) {
  (void)in_sizes; (void)n_in; (void)out_size; (void)ws_size;
  const float* x      = (const float*)d_in[0];   // [16384, 512]
  const float* angles = (const float*)d_in[1];   // [130816]
  const float* bias   = (const float*)d_in[2];   // [512]
  float* out = (float*)d_out;                    // [16384, 512]

  const size_t MAT = (size_t)DIM * DIM;          // 1 MB fp32 each
  float* A  = (float*)d_ws;
  float* W  = A + MAT;
  float* T0 = W + MAT;
  float* T1 = T0 + MAT;

  // 1) A from angles
  build_skew_kernel<<<dim3(DIM / 64, DIM / 4), 256, 0, stream>>>(angles, A);
  // 2) W = I + A
  init_w_kernel<<<(int)(MAT / 256), 256, 0, stream>>>(A, W);
  // 3) Taylor: T_k = T_{k-1} @ A / k ;  W += T_k   (128x64 tiles -> 32 blocks)
  const float* Tprev = A;
  float* pp[2] = {T0, T1};
  int b = 0;
  for (int k = 2; k <= N_TAYLOR; ++k) {
    float* Tnew = pp[b]; b ^= 1;
    gemm_f32_wmma_kernel<2><<<dim3(DIM / 64, DIM / BM), 256, 0, stream>>>(
        Tprev, A, Tnew, 1.0f / (float)k, W, nullptr);
    Tprev = Tnew;
  }
  // 4) out = x @ W + bias   (128x128 tiles -> 512 blocks)
  gemm_f32_wmma_kernel<4><<<dim3(DIM / 128, 16384 / BM), 256, 0, stream>>>(
      x, W, out, 1.0f, nullptr, bias);
}